// IWHT3Layer_15522011808175
// MI455X (gfx1250) — compile-verified
//
#include <hip/hip_runtime.h>

typedef float v2f __attribute__((ext_vector_type(2)));
typedef float v4f __attribute__((ext_vector_type(4)));
typedef float v8f __attribute__((ext_vector_type(8)));

#define NPOS      12544            // 16*28*28 sites per branch per t
#define CDIM      64
#define KDIM      64
#define MTILE     32               // sites per workgroup (two 16-row WMMA M-tiles)
#define LDSTRIDE  68               // padded row stride (floats) -> bank-conflict-free
#define OUT_PER_BRANCH (16*112*112*64)

__global__ __launch_bounds__(256) void iwht3_fused_kernel(
    const float* __restrict__ t1, const float* __restrict__ t2, const float* __restrict__ t3,
    const float* __restrict__ w1, const float* __restrict__ w2, const float* __restrict__ w3,
    const float* __restrict__ b1, const float* __restrict__ b2, const float* __restrict__ b3,
    float* __restrict__ out)
{
    __shared__ float smem[16 * MTILE * LDSTRIDE];   // 139,264 B; A tile, then reused for y

    const int branch = blockIdx.y;
    const float* trans = (branch == 0) ? t1 : (branch == 1) ? t2 : t3;
    const float* wgt   = (branch == 0) ? w1 : (branch == 1) ? w2 : w3;
    const float* bias  = (branch == 0) ? b1 : (branch == 1) ? b2 : b3;
    float* outp = out + (size_t)branch * OUT_PER_BRANCH;

    const int tile = blockIdx.x;       // 0..391
    const int pos0 = tile * MTILE;     // 32 consecutive flattened sites (contiguous in memory)

    const int tid = threadIdx.x;

    // ---- Stage 1: cooperative load of A (16 t-slices, each contiguous 8KB) into LDS ----
    for (int t = 0; t < 16; ++t) {
        #pragma unroll
        for (int pass = 0; pass < 2; ++pass) {
            const int idx = pass * 256 + tid;       // float4 index within t-slice (0..511)
            const int m = idx >> 4;                 // 0..31 : site within tile
            const int c = (idx & 15) << 2;          // 0,4,...,60
            const float* g = trans + ((size_t)(t * NPOS + pos0 + m)) * CDIM + c;
            v4f v = *(const v4f*)g;                            // coalesced global_load_b128
            *(v4f*)&smem[(t * MTILE + m) * LDSTRIDE + c] = v;  // ds_store_b128
        }
    }
    __syncthreads();

    const int lane = tid & 31;
    const int wv   = tid >> 5;     // wave 0..7 ; owns t = 2*wv, 2*wv+1
    const int hl   = lane >> 4;    // half-wave: 0 -> K pair {0,1}, 1 -> {2,3}
    const int ml   = lane & 15;

    // ---- Stage 2: GEMM via V_WMMA_F32_16X16X4_F32, B fragments reused across 2 M-tiles ----
    v8f acc[2][4][2];
    #pragma unroll
    for (int tt = 0; tt < 2; ++tt)
        #pragma unroll
        for (int n = 0; n < 4; ++n)
            #pragma unroll
            for (int mt = 0; mt < 2; ++mt) {
                v8f z = {0.f,0.f,0.f,0.f,0.f,0.f,0.f,0.f};
                acc[tt][n][mt] = z;
            }

    #pragma unroll
    for (int tt = 0; tt < 2; ++tt) {
        const int t = wv * 2 + tt;
        const float* wb = wgt + (size_t)t * CDIM * KDIM;             // [C][K], K contiguous
        const float* aBase0 = &smem[(t * MTILE + ml) * LDSTRIDE + 2 * hl];
        const float* aBase1 = aBase0 + 16 * LDSTRIDE;
        for (int ks = 0; ks < 16; ++ks) {                            // K in steps of 4
            v2f a0 = *(const v2f*)(aBase0 + ks * 4);                 // ds_load_b64
            v2f a1 = *(const v2f*)(aBase1 + ks * 4);
            const int c0 = ks * 4 + 2 * hl;
            #pragma unroll
            for (int n = 0; n < 4; ++n) {
                v2f b;
                b.x = wb[(size_t)c0 * KDIM + n * 16 + ml];           // B[c0][k]
                b.y = wb[(size_t)(c0 + 1) * KDIM + n * 16 + ml];     // B[c0+1][k]
                acc[tt][n][0] = __builtin_amdgcn_wmma_f32_16x16x4_f32(
                    false, a0, false, b, (short)0, acc[tt][n][0], false, false);
                acc[tt][n][1] = __builtin_amdgcn_wmma_f32_16x16x4_f32(
                    false, a1, false, b, (short)0, acc[tt][n][1], false, false);
            }
        }
    }

    // ---- Stage 3: spill y[t][m][k] into LDS (reuse A buffer) ----
    __syncthreads();   // all waves done reading A
    #pragma unroll
    for (int tt = 0; tt < 2; ++tt) {
        const int t = wv * 2 + tt;
        #pragma unroll
        for (int n = 0; n < 4; ++n)
            #pragma unroll
            for (int mt = 0; mt < 2; ++mt)
                #pragma unroll
                for (int r = 0; r < 8; ++r) {
                    const int m = mt * 16 + r + 8 * hl;   // C/D layout: M = r + 8*half
                    smem[(t * MTILE + m) * LDSTRIDE + n * 16 + ml] = acc[tt][n][mt][r];
                }
    }
    __syncthreads();

    // ---- Stage 4: inverse 2-D WHT (fast butterflies) + bias, scatter 4x4 blocks ----
    const int k  = tid & 63;       // output channel
    const int mg = tid >> 6;       // 0..3 -> handles sites mg*8 .. mg*8+7
    const float bk = bias[k];

    for (int ii = 0; ii < 8; ++ii) {
        const int m = mg * 8 + ii;

        float y[4][4];
        #pragma unroll
        for (int u = 0; u < 4; ++u)
            #pragma unroll
            for (int v = 0; v < 4; ++v)
                y[u][v] = smem[((u * 4 + v) * MTILE + m) * LDSTRIDE + k];

        float w[4][4];
        #pragma unroll
        for (int v = 0; v < 4; ++v) {
            const float a = y[0][v] + y[2][v], c = y[0][v] - y[2][v];
            const float b = y[1][v] + y[3][v], d = y[1][v] - y[3][v];
            w[0][v] = a + b; w[1][v] = a - b; w[2][v] = c + d; w[3][v] = c - d;
        }

        const int pos = pos0 + m;
        const int bb  = pos / 784;
        const int ij  = pos % 784;
        const int i   = ij / 28, j = ij % 28;
        float* obase = outp + ((size_t)(bb * 112 + i * 4) * 112 + (size_t)(j * 4)) * KDIM + k;

        #pragma unroll
        for (int p = 0; p < 4; ++p) {
            const float a = w[p][0] + w[p][2], c = w[p][0] - w[p][2];
            const float b = w[p][1] + w[p][3], d = w[p][1] - w[p][3];
            float* r = obase + (size_t)p * 112 * KDIM;
            r[0 * KDIM] = (a + b) * 0.0625f + bk;
            r[1 * KDIM] = (a - b) * 0.0625f + bk;
            r[2 * KDIM] = (c + d) * 0.0625f + bk;
            r[3 * KDIM] = (c - d) * 0.0625f + bk;
        }
    }
}

extern "C" void kernel_launch(void* const* d_in, const int* in_sizes, int n_in,
                              void* d_out, int out_size, void* d_ws, size_t ws_size,
                              hipStream_t stream) {
    (void)in_sizes; (void)n_in; (void)out_size; (void)d_ws; (void)ws_size;
    const float* t1 = (const float*)d_in[0];
    const float* t2 = (const float*)d_in[1];
    const float* t3 = (const float*)d_in[2];
    const float* w1 = (const float*)d_in[3];
    const float* w2 = (const float*)d_in[4];
    const float* w3 = (const float*)d_in[5];
    const float* b1 = (const float*)d_in[6];
    const float* b2 = (const float*)d_in[7];
    const float* b3 = (const float*)d_in[8];
    float* out = (float*)d_out;

    dim3 grid(392, 3, 1);   // 392 x 32-site tiles per branch, 3 branches
    dim3 block(256, 1, 1);  // 8 waves (wave32): one t-pair per wave, 2 M-tiles each
    iwht3_fused_kernel<<<grid, block, 0, stream>>>(t1, t2, t3, w1, w2, w3, b1, b2, b3, out);
}